// GradientLoss_42537356099619
// MI455X (gfx1250) — compile-verified
//
#include <hip/hip_runtime.h>
#include <math.h>

// Problem constants
#define BATCH 32
#define CHANS 3
#define HT    512
#define WD    512
#define PLANE (HT * WD)           // 262144
#define ROWS  32                  // rows per workgroup tile (divides 512)
#define TROWS (ROWS + 2)          // with halo
#define NBLK  16                  // 4x4 loss blocks per image

// --- CDNA5 async global->LDS path (gfx1250), with compile-safe fallback ---
#if defined(__has_builtin)
#  if __has_builtin(__builtin_amdgcn_global_load_async_to_lds_b128)
#    define USE_ASYNC_LDS 1
#  endif
#  if __has_builtin(__builtin_amdgcn_s_wait_asynccnt)
#    define HAVE_WAIT_ASYNCCNT 1
#  endif
#endif

// Pointee type per clang's builtin signature: vector of 4 ints (16 bytes).
typedef int v4i __attribute__((vector_size(16)));
typedef __attribute__((address_space(1))) v4i gv4i;   // global
typedef __attribute__((address_space(3))) v4i lv4i;   // LDS

__device__ __forceinline__ void async_copy16(const float4* gsrc, float4* ldst) {
#if defined(USE_ASYNC_LDS)
    __builtin_amdgcn_global_load_async_to_lds_b128(
        (gv4i*)gsrc, (lv4i*)ldst, 0, 0);
#else
    *ldst = *gsrc;
#endif
}

__device__ __forceinline__ void async_wait_all() {
#if defined(USE_ASYNC_LDS)
#  if defined(HAVE_WAIT_ASYNCCNT)
    __builtin_amdgcn_s_wait_asynccnt(0);
#  else
    asm volatile("s_wait_asynccnt 0" ::: "memory");
#  endif
#endif
}

// Zero the 32*16 block accumulators in workspace (graph-capture safe).
__global__ void GL_zero_ws(float* __restrict__ ws) {
    ws[threadIdx.x] = 0.0f;
}

// Main kernel: one workgroup = 32 rows of one (b, c) plane.
// Stage (ROWS+2)x512 of both inputs into LDS via async copies, then each wave
// sweeps a 16-row x 128-col region with a rolling 3-row register window so
// every LDS element is read ~once (ds_load_b128 per lane per row).
__global__ void __launch_bounds__(256)
GL_lap_blocks(const float* __restrict__ gt, const float* __restrict__ sp,
              float* __restrict__ ws) {
    __shared__ float G[TROWS * WD];
    __shared__ float S[TROWS * WD];
    __shared__ float acc[NBLK];

    const int tid = threadIdx.x;
    const int b  = blockIdx.z;
    const int c  = blockIdx.y;
    const int r0 = blockIdx.x * ROWS;

    if (tid < NBLK) acc[tid] = 0.0f;

    const float* gbase = gt + ((size_t)b * CHANS + c) * PLANE;
    const float* sbase = sp + ((size_t)b * CHANS + c) * PLANE;

    // ---- Stage tile (ROWS+2 rows x 512 cols) of both inputs into LDS ----
    for (int m = tid; m < TROWS * (WD / 4); m += 256) {
        const int k  = m >> 7;        // tile row 0..33
        const int c4 = m & 127;       // float4 column
        int gr = r0 - 1 + k;          // global row with reflect padding
        gr = (gr < 0) ? -gr : ((gr > HT - 1) ? 2 * (HT - 1) - gr : gr);
        const float4* gs = (const float4*)(gbase + (size_t)gr * WD) + c4;
        const float4* ss = (const float4*)(sbase + (size_t)gr * WD) + c4;
        async_copy16(gs, (float4*)&G[k * WD + c4 * 4]);
        async_copy16(ss, (float4*)&S[k * WD + c4 * 4]);
    }
    async_wait_all();
    __syncthreads();

    // ---- Compute: rolling-window Laplacian diff + per-row block add ----
    const int wave = tid >> 5;
    const int lane = tid & 31;
    const int p       = wave >> 1 >> 1;     // wave/4: row half (0/1)
    const int cg      = wave & 3;           // 128-col group 0..3
    const int colbase = cg << 7;            // 0,128,256,384
    const int c0      = colbase + (lane << 2);  // lane's 4-col strip

    float4 hsA, hsB, hsC;   // hsum window: rows lr-2, lr-1, lr
    float4 tPrev;           // t at row lr-1 (center of output row)
    hsA = hsB = hsC = make_float4(0.f, 0.f, 0.f, 0.f);
    tPrev = hsA;

    float myAcc[1]; (void)myAcc;
    for (int k = 0; k < 18; ++k) {
        const int lr = (p << 4) + k;        // local tile row 0..33
        const int off = lr * WD + c0;
        const float4 g4 = *(const float4*)&G[off];
        const float4 s4 = *(const float4*)&S[off];
        float4 t;
        t.x = g4.x - s4.x; t.y = g4.y - s4.y;
        t.z = g4.z - s4.z; t.w = g4.w - s4.w;

        // strip-edge neighbors across lanes (wave32 shuffles)
        float tL = __shfl_up(t.w, 1, 32);
        float tR = __shfl_down(t.x, 1, 32);
        if (lane == 0) {
            tL = (colbase == 0) ? t.y                      // reflect col -1 -> 1
                 : (G[lr * WD + colbase - 1] - S[lr * WD + colbase - 1]);
        }
        if (lane == 31) {
            tR = (colbase == 384) ? t.z                    // reflect col 512 -> 510
                 : (G[lr * WD + colbase + 128] - S[lr * WD + colbase + 128]);
        }

        float4 hs;
        hs.x = tL  + t.x + t.y;
        hs.y = t.x + t.y + t.z;
        hs.z = t.y + t.z + t.w;
        hs.w = t.z + t.w + tR;

        if (k >= 2) {
            // output local row lr-1 -> global row h
            const int h  = r0 + (p << 4) + k - 2;
            const float4 vx = hsB;   // hs[h] (center row hsum)
            float d0 = fabsf((hsA.x + vx.x + hs.x - 9.0f * tPrev.x) * 0.0625f);
            float d1 = fabsf((hsA.y + vx.y + hs.y - 9.0f * tPrev.y) * 0.0625f);
            float d2 = fabsf((hsA.z + vx.z + hs.z - 9.0f * tPrev.z) * 0.0625f);
            float d3 = fabsf((hsA.w + vx.w + hs.w - 9.0f * tPrev.w) * 0.0625f);
            float rsum = (d0 + d1) + (d2 + d3);
            #pragma unroll
            for (int off2 = 16; off2 > 0; off2 >>= 1)
                rsum += __shfl_xor(rsum, off2, 32);
            if (lane == 0) {
                const int v  = (512 * c + h) / 384;        // vertical block
                const int m3 = (2 * c + h) % 3;            // flat phase
                const int hz = ((m3 << 9) + colbase) / 384; // whole group -> one hz
                atomicAdd(&acc[(v << 2) + hz], rsum);      // ds_add_f32
            }
        }
        hsA = hsB; hsB = hs; tPrev = t;
        (void)hsC;
    }
    __syncthreads();

    if (tid < NBLK) atomicAdd(&ws[b * NBLK + tid], acc[tid]);
}

// Final: per-image max over 16 blocks, sum over 32 images -> out[0]. One wave.
__global__ void GL_finalize(const float* __restrict__ ws, float* __restrict__ out) {
    const int t = threadIdx.x;  // 0..31 = image index
    float m = ws[t * NBLK];
    #pragma unroll
    for (int i = 1; i < NBLK; ++i) m = fmaxf(m, ws[t * NBLK + i]);
    #pragma unroll
    for (int off = 16; off > 0; off >>= 1) m += __shfl_xor(m, off, 32);
    if (t == 0) out[0] = m;
}

extern "C" void kernel_launch(void* const* d_in, const int* in_sizes, int n_in,
                              void* d_out, int out_size, void* d_ws, size_t ws_size,
                              hipStream_t stream) {
    (void)in_sizes; (void)n_in; (void)out_size; (void)ws_size;
    const float* gt = (const float*)d_in[0];  // ground_truth_batch
    const float* sp = (const float*)d_in[1];  // segmented_print_batch
    float* out = (float*)d_out;
    float* ws  = (float*)d_ws;                // 32*16 block accumulators

    GL_zero_ws<<<1, BATCH * NBLK, 0, stream>>>(ws);
    dim3 grid(HT / ROWS, CHANS, BATCH);       // (16, 3, 32) workgroups
    GL_lap_blocks<<<grid, 256, 0, stream>>>(gt, sp, ws);
    GL_finalize<<<1, 32, 0, stream>>>(ws, out);
}